// GGG_34033320853960
// MI455X (gfx1250) — compile-verified
//
#include <hip/hip_runtime.h>
#include <hip/hip_bf16.h>
#include <math.h>

typedef __attribute__((ext_vector_type(2))) float v2f;
typedef __attribute__((ext_vector_type(8))) float v8f;

#define HIDD   768
#define HID2   1536
#define NHEAD  12
#define BTOK   8192      // B*T
#define TLEN   1024

// ---------------------------------------------------------------------------
// Kernel 1: max-pool (window 7, pad 3) along feature dim + concat
// Acat[m, 0:768] = maxpool(x[m,:]), Acat[m, 768:1536] = x[m,:]
// ---------------------------------------------------------------------------
__global__ __launch_bounds__(256)
void pool_concat_kernel(const float* __restrict__ x, float* __restrict__ acat) {
    __shared__ float row[HIDD];
    const int m = blockIdx.x;
    const float* xr = x + (size_t)m * HIDD;
    float* ar = acat + (size_t)m * HID2;
    for (int h = threadIdx.x; h < HIDD; h += 256) row[h] = xr[h];
    __syncthreads();
    for (int h = threadIdx.x; h < HIDD; h += 256) {
        int lo = h - 3 < 0 ? 0 : h - 3;
        int hq = h + 3 > HIDD - 1 ? HIDD - 1 : h + 3;
        float mx = row[lo];
        for (int q = lo + 1; q <= hq; ++q) mx = fmaxf(mx, row[q]);
        ar[h] = mx;
        ar[HIDD + h] = row[h];
    }
}

// ---------------------------------------------------------------------------
// Kernel 2: x_new = Acat @ Wt + bt via V_WMMA_F32_16X16X4_F32
// M=8192, K=1536, N=768. Block = 4 waves; each wave computes a 32x64 tile.
// A-frag  (16x4 f32): lanes 0-15 -> M rows, vgpr pair holds K={0,1} (lo half)
//                     or K={2,3} (hi half of wave).
// B-frag  (4x16 f32): lanes 0-15 -> N cols of row K, lo half rows {k,k+1},
//                     hi half rows {k+2,k+3}.
// D tile (16x16 f32): vgpr r: lanes0-15 M=r, lanes16-31 M=r+8.
// ---------------------------------------------------------------------------
__global__ __launch_bounds__(128)
void gemm1_wmma_kernel(const float* __restrict__ A, const float* __restrict__ Wt,
                       const float* __restrict__ bt, float* __restrict__ C) {
    const int lane  = threadIdx.x & 31;
    const int wave  = threadIdx.x >> 5;
    const int laneM = lane & 15;
    const int hi    = lane >> 4;          // 0: lanes 0-15, 1: lanes 16-31
    const int m_base = blockIdx.y * 128 + wave * 32;
    const int n_base = blockIdx.x * 64;

    const v8f vzero = {0.f, 0.f, 0.f, 0.f, 0.f, 0.f, 0.f, 0.f};
    v8f acc[2][4];
#pragma unroll
    for (int s = 0; s < 2; ++s)
#pragma unroll
        for (int nt = 0; nt < 4; ++nt) acc[s][nt] = vzero;

    const float* arow0 = A + (size_t)(m_base + laneM) * HID2;
    const float* arow1 = arow0 + (size_t)16 * HID2;

    for (int k = 0; k < HID2; k += 4) {
        const int kk = k + 2 * hi;
        v2f a0 = *(const v2f*)(arow0 + kk);          // rows m_base+0..15
        v2f a1 = *(const v2f*)(arow1 + kk);          // rows m_base+16..31
        const float* bp = Wt + (size_t)kk * HIDD + n_base + laneM;
#pragma unroll
        for (int nt = 0; nt < 4; ++nt) {
            v2f b;
            b.x = bp[nt * 16];                        // row kk
            b.y = bp[nt * 16 + HIDD];                 // row kk+1
            acc[0][nt] = __builtin_amdgcn_wmma_f32_16x16x4_f32(
                false, a0, false, b, (short)0, acc[0][nt], false, false);
            acc[1][nt] = __builtin_amdgcn_wmma_f32_16x16x4_f32(
                false, a1, false, b, (short)0, acc[1][nt], false, false);
        }
    }

#pragma unroll
    for (int s = 0; s < 2; ++s) {
#pragma unroll
        for (int nt = 0; nt < 4; ++nt) {
            const int col  = n_base + nt * 16 + laneM;
            const float bb = bt[col];
#pragma unroll
            for (int r = 0; r < 8; ++r) {
                const int rowm = m_base + s * 16 + r + 8 * hi;
                C[(size_t)rowm * HIDD + col] = acc[s][nt][r] + bb;
            }
        }
    }
}

// ---------------------------------------------------------------------------
// Kernel 3: gp = x_new @ Wg + bg, then per-token stats:
//   mean[h], s2 = std^2, L = log(std), inv = 1/std^2   (std = softplus)
// Block = 4 waves, one row per wave; lanes 0-23 compute the 24 dots.
// ---------------------------------------------------------------------------
__global__ __launch_bounds__(128)
void gp_stats_kernel(const float* __restrict__ xn, const float* __restrict__ Wg,
                     const float* __restrict__ bg,
                     float* __restrict__ meanA, float* __restrict__ s2A,
                     float* __restrict__ logA,  float* __restrict__ invA) {
    __shared__ float rows[4][HIDD];
    const int m0 = blockIdx.x * 4;
    for (int idx = threadIdx.x; idx < 4 * HIDD; idx += 128)
        rows[idx / HIDD][idx % HIDD] = xn[(size_t)m0 * HIDD + idx];
    __syncthreads();

    const int wave = threadIdx.x >> 5;
    const int lane = threadIdx.x & 31;
    const int m = m0 + wave;
    if (lane < 24) {
        float acc = bg[lane];
        for (int k = 0; k < HIDD; ++k)
            acc = fmaf(rows[wave][k], Wg[k * 24 + lane], acc);
        if (lane < NHEAD) {
            meanA[m * NHEAD + lane] = acc;
        } else {
            const int h = lane - NHEAD;
            // softplus, numerically stable
            float sp = acc > 20.f ? acc : log1pf(expf(acc));
            s2A[m * NHEAD + h]  = sp * sp;
            logA[m * NHEAD + h] = logf(sp);
            invA[m * NHEAD + h] = 1.f / (sp * sp);
        }
    }
}

// ---------------------------------------------------------------------------
// Kernel 4: kl[b,i,j,h] = 0.5*(2*L_i - 2*L_j + (S2_j + (M_j - M_i)^2)*inv_i - 1)
// One block per (b,i); streaming-store bound (403 MB output).
// ---------------------------------------------------------------------------
__global__ __launch_bounds__(256)
void kl_kernel(const float* __restrict__ meanA, const float* __restrict__ s2A,
               const float* __restrict__ logA,  const float* __restrict__ invA,
               float* __restrict__ kl) {
    __shared__ float Mi[NHEAD], Li2[NHEAD], Ii[NHEAD];
    const int b = blockIdx.y;
    const int i = blockIdx.x;
    const int base_i = (b * TLEN + i) * NHEAD;
    if (threadIdx.x < NHEAD) {
        Mi[threadIdx.x]  = meanA[base_i + threadIdx.x];
        Li2[threadIdx.x] = 2.f * logA[base_i + threadIdx.x];
        Ii[threadIdx.x]  = invA[base_i + threadIdx.x];
    }
    __syncthreads();

    const float* mj  = meanA + (size_t)b * TLEN * NHEAD;
    const float* s2j = s2A   + (size_t)b * TLEN * NHEAD;
    const float* lj  = logA  + (size_t)b * TLEN * NHEAD;
    float* out = kl + ((size_t)(b * TLEN + i)) * TLEN * NHEAD;

    for (int f = threadIdx.x; f < TLEN * NHEAD; f += 256) {
        const int h = f % NHEAD;
        const float d = mj[f] - Mi[h];
        out[f] = 0.5f * (Li2[h] - 2.f * lj[f] + (s2j[f] + d * d) * Ii[h] - 1.f);
    }
}

// ---------------------------------------------------------------------------
extern "C" void kernel_launch(void* const* d_in, const int* in_sizes, int n_in,
                              void* d_out, int out_size, void* d_ws, size_t ws_size,
                              hipStream_t stream) {
    const float* x  = (const float*)d_in[0];
    // d_in[1] = adj (unused by the reference)
    const float* Wt = (const float*)d_in[2];
    const float* bt = (const float*)d_in[3];
    const float* Wg = (const float*)d_in[4];
    const float* bg = (const float*)d_in[5];

    float* xnew = (float*)d_out;                          // 8192*768
    float* kl   = xnew + (size_t)BTOK * HIDD;             // 8*1024*1024*12

    float* acat  = (float*)d_ws;                          // 8192*1536
    float* meanA = acat + (size_t)BTOK * HID2;            // 8192*12 each
    float* s2A   = meanA + (size_t)BTOK * NHEAD;
    float* logA  = s2A   + (size_t)BTOK * NHEAD;
    float* invA  = logA  + (size_t)BTOK * NHEAD;

    pool_concat_kernel<<<BTOK, 256, 0, stream>>>(x, acat);

    dim3 g1(HIDD / 64, BTOK / 128);                       // (12, 64)
    gemm1_wmma_kernel<<<g1, 128, 0, stream>>>(acat, Wt, bt, xnew);

    gp_stats_kernel<<<BTOK / 4, 128, 0, stream>>>(xnew, Wg, bg,
                                                  meanA, s2A, logA, invA);

    dim3 g2(TLEN, 8);
    kl_kernel<<<g2, 256, 0, stream>>>(meanA, s2A, logA, invA, kl);
}